// RecurrentAccumulator_81707457839234
// MI455X (gfx1250) — compile-verified
//
#include <hip/hip_runtime.h>
#include <stdint.h>

typedef __attribute__((ext_vector_type(2))) float v2f;
typedef __attribute__((ext_vector_type(8))) float v8f;

// Force saddr-form store: global_store_b32 voffset, vdata, s[base]
// (base uniform per time-slice; voffset is the shared per-lane byte offset).
__device__ __forceinline__ void store_saddr(const char* base, uint32_t voff, float v) {
    asm volatile("global_store_b32 %0, %1, %2"
                 :
                 : "v"(voff), "v"(v), "s"(base));
}

// cummean over T=8 as a lower-triangular matmul on the WMMA unit:
//   D(16x16) = Wlo(16x8) x XtileA(8x16) + Whi(16x8) x XtileB(8x16)
// Wlo rows 0-7 = W (W[t][s] = 1/(t+1), s<=t), rows 8-15 = 0.
// Whi rows 0-7 = 0, rows 8-15 = W.
// => D VGPR t: lanes 0-15 = out[t] cols base..+15 (tile A),
//              lanes 16-31 = out[t] cols base+16..+31 (tile B)
// so every output VGPR is one contiguous 128B store.
// K=8 split as 2x K=4 per tile => 4 chained V_WMMA_F32_16X16X4_F32 / 32 cols,
// full f32 precision end to end.
__global__ __launch_bounds__(256)
void cummean_wmma_kernel(const float* __restrict__ x, float* __restrict__ out,
                         long long N,        // columns per time slice (for tail)
                         long long tiles,    // N / 32
                         uint32_t sliceB,    // N * 4 bytes
                         uint32_t strideB,   // wavesTotal * 128 bytes
                         int itersBase,      // tiles / wavesTotal
                         int itersRem) {     // tiles % wavesTotal
    const int lane = threadIdx.x & 31;
    const long long wave =
        ((long long)blockIdx.x * (long long)blockDim.x + (long long)threadIdx.x) >> 5;

    const int n  = lane & 15;          // column within 16-wide tile / A-row M
    const int kh = (lane >> 4) << 1;   // K-half: 0 (lanes 0-15) or 2 (lanes 16-31)
    const int m  = n;                  // A-matrix row held by this lane
    const int r  = m & 7;              // logical time row (0..7)

    // One-time weight setup (precise divide once per wave, hoisted).
    const float inv   = 1.0f / (float)(r + 1);
    const float selLo = (m < 8) ? inv : 0.0f;    // rows 0-7 active (tile A)
    const float selHi = (m < 8) ? 0.0f : inv;    // rows 8-15 active (tile B)
    v2f a0, a1, a0h, a1h;                        // A chunks: K = kh+{0,1}, kh+4+{0,1}
    a0.x  = ((kh + 0) <= r) ? selLo : 0.0f;
    a0.y  = ((kh + 1) <= r) ? selLo : 0.0f;
    a1.x  = ((kh + 4) <= r) ? selLo : 0.0f;
    a1.y  = ((kh + 5) <= r) ? selLo : 0.0f;
    a0h.x = ((kh + 0) <= r) ? selHi : 0.0f;
    a0h.y = ((kh + 1) <= r) ? selHi : 0.0f;
    a1h.x = ((kh + 4) <= r) ? selHi : 0.0f;
    a1h.y = ((kh + 5) <= r) ? selHi : 0.0f;

    // Uniform SGPR bases + per-lane 32-bit byte offsets (tensor = 411MB < 4GB)
    // => saddr-form global loads AND stores, cheap u32 increments per iteration.
    const char* __restrict__ xb = (const char*)x;
    const char* ob = (const char*)out;
    const char* const o0 = ob;
    const char* const o1 = ob + 1 * (size_t)sliceB;
    const char* const o2 = ob + 2 * (size_t)sliceB;
    const char* const o3 = ob + 3 * (size_t)sliceB;
    const char* const o4 = ob + 4 * (size_t)sliceB;
    const char* const o5 = ob + 5 * (size_t)sliceB;
    const char* const o6 = ob + 6 * (size_t)sliceB;
    const char* const o7 = ob + 7 * (size_t)sliceB;

    const uint32_t col0 = (uint32_t)(wave * 32);
    // B matrix (4x16 f32): VGPR j, lane L -> (K = j + 2*(L>=16), N = L%16).
    // This lane feeds slices kh+{0,1,4,5}; tile B = same offsets + imm 64.
    uint32_t l0 = (uint32_t)kh * sliceB + (col0 + (uint32_t)n) * 4u;
    uint32_t l1 = l0 + sliceB;        // slice kh+1
    uint32_t l4 = l0 + 4u * sliceB;   // slice kh+4
    uint32_t l5 = l0 + 5u * sliceB;   // slice kh+5
    uint32_t so = (col0 + (uint32_t)lane) * 4u;   // shared store offset

    // Scalar (SGPR) trip count -> pure s_cmp/s_cbranch loop, EXEC stays all-ones
    // around WMMA as the ISA requires. iters is wave-uniform by construction.
    int iters = itersBase + ((wave < (long long)itersRem) ? 1 : 0);
    iters = __builtin_amdgcn_readfirstlane(iters);

    for (int i = 0; i < iters; ++i) {
        v2f b0, b1, b0h, b1h;
        b0.x  = *(const float*)(xb + l0);
        b0.y  = *(const float*)(xb + l1);
        b1.x  = *(const float*)(xb + l4);
        b1.y  = *(const float*)(xb + l5);
        b0h.x = *(const float*)(xb + l0 + 64);
        b0h.y = *(const float*)(xb + l1 + 64);
        b1h.x = *(const float*)(xb + l4 + 64);
        b1h.y = *(const float*)(xb + l5 + 64);

        v8f c = {};
        c = __builtin_amdgcn_wmma_f32_16x16x4_f32(false, a0,  false, b0,  (short)0, c, false, false);
        c = __builtin_amdgcn_wmma_f32_16x16x4_f32(false, a1,  false, b1,  (short)0, c, false, false);
        c = __builtin_amdgcn_wmma_f32_16x16x4_f32(false, a0h, false, b0h, (short)0, c, false, false);
        c = __builtin_amdgcn_wmma_f32_16x16x4_f32(false, a1h, false, b1h, (short)0, c, false, false);

        // 8 contiguous 128B stores: uniform SGPR base per time step + shared voffset.
        store_saddr(o0, so, c[0]);
        store_saddr(o1, so, c[1]);
        store_saddr(o2, so, c[2]);
        store_saddr(o3, so, c[3]);
        store_saddr(o4, so, c[4]);
        store_saddr(o5, so, c[5]);
        store_saddr(o6, so, c[6]);
        store_saddr(o7, so, c[7]);

        l0 += strideB; l1 += strideB; l4 += strideB; l5 += strideB; so += strideB;
    }

    // Remainder columns (N % 32 != 0): scalar path, wave-uniform guard, no WMMA.
    const long long rem0 = tiles * 32;
    if (wave == 0 && rem0 < N) {
        const long long col = rem0 + lane;
        if (col < N) {
            float s = 0.0f;
            #pragma unroll
            for (int t = 0; t < 8; ++t) {
                s += x[(long long)t * N + col];
                out[(long long)t * N + col] = s / (float)(t + 1);
            }
        }
    }
}

extern "C" void kernel_launch(void* const* d_in, const int* in_sizes, int n_in,
                              void* d_out, int out_size, void* d_ws, size_t ws_size,
                              hipStream_t stream) {
    const float* x = (const float*)d_in[0];
    float* out = (float*)d_out;

    const long long total = (long long)in_sizes[0];   // 8 * 64 * 256 * 28 * 28
    const long long T = 8;
    const long long N = total / T;                    // 12,845,056 columns per slice
    const long long tiles = N / 32;                   // 401,408 two-tile groups

    // ~16 tiles per wave: plenty of waves for occupancy, enough iterations to
    // amortize setup. Whole 256-thread blocks (8 waves each).
    long long wavesWanted = (tiles + 15) / 16;
    if (wavesWanted < 1) wavesWanted = 1;
    long long blocks = (wavesWanted + 7) / 8;
    long long wavesTotal = blocks * 8;

    const uint32_t sliceB  = (uint32_t)(N * 4);
    const uint32_t strideB = (uint32_t)(wavesTotal * 128);
    const int itersBase = (int)(tiles / wavesTotal);
    const int itersRem  = (int)(tiles % wavesTotal);

    cummean_wmma_kernel<<<(unsigned)blocks, 256, 0, stream>>>(
        x, out, N, tiles, sliceB, strideB, itersBase, itersRem);
}